// CircuitLayer_3075196584637
// MI455X (gfx1250) — compile-verified
//
#include <hip/hip_runtime.h>

// ---------------------------------------------------------------------------
// CircuitLayer edge-model for MI455X (gfx1250, wave32).
//
// aux[(N+1)][16] node-major table (row 0 = phantom zero node) built with
// V_WMMA_F32_16X16X4_F32 identity-slice transposes (bit-exact: B is one-hot),
// edge gather/tanh/scatter with one edge per 16-lane half-wave (contiguous
// 64B gathers + coalesced global_atomic_add_f32 into the L2-resident
// accumulator), then WMMA-transpose back to (B, N).
//
// Full 16x16 tiles take a branch-free WMMA path (EXEC all-ones, clause-able
// load bursts); any tail columns (none for N=50000) go through a scalar
// kernel so the WMMA path never needs per-load predication.
// ---------------------------------------------------------------------------

typedef __attribute__((ext_vector_type(2))) float v2f;
typedef __attribute__((ext_vector_type(8))) float v8f;

// B_p identity slice for V_WMMA_F32_16X16X4_F32 (B is 4x16, 2 VGPRs):
//   VGPR0: lanes 0-15 -> K=0, lanes 16-31 -> K=2
//   VGPR1: lanes 0-15 -> K=1, lanes 16-31 -> K=3
//   B_p[k][n] = (n == 4p + k)
__device__ __forceinline__ v2f id_slice(int lane, int p) {
  v2f b;
  b.x = (lane == 4 * p     || lane == 4 * p + 18) ? 1.0f : 0.0f; // K=0 / K=2
  b.y = (lane == 4 * p + 1 || lane == 4 * p + 19) ? 1.0f : 0.0f; // K=1 / K=3
  return b;
}

__global__ void zero_kernel(float* __restrict__ p, long n) {
  long i = (long)blockIdx.x * blockDim.x + threadIdx.x;
  long stride = (long)gridDim.x * blockDim.x;
  for (; i < n; i += stride) p[i] = 0.0f;
}

// x (16 x N row-major) -> aux ((N+1) x 16): aux[n+1][b] = x[b][n].
// Full tiles only; one wave per tile; no bounds checks (EXEC all-ones).
__global__ void xpose_fwd_kernel(const float* __restrict__ x,
                                 float* __restrict__ aux,
                                 int N, int ntiles_full) {
  const int lane = threadIdx.x & 31;
  const int wid  = threadIdx.x >> 5;
  const int wpb  = blockDim.x >> 5;
  const int m    = lane & 15;          // A-layout M index == batch lane
  const bool hi  = lane >= 16;
  const int k0   = hi ? 2 : 0;

  for (int t = blockIdx.x * wpb + wid; t < ntiles_full; t += gridDim.x * wpb) {
    const int col0 = t << 4;
    const float* __restrict__ colp = x + col0 + m;   // lane's column pointer
    v8f acc = {};
#pragma unroll
    for (int p = 0; p < 4; ++p) {
      v2f a;
      a.x = colp[(long)(4 * p + k0)     * N];
      a.y = colp[(long)(4 * p + k0 + 1) * N];
      v2f b = id_slice(lane, p);
      acc = __builtin_amdgcn_wmma_f32_16x16x4_f32(
          false, a, false, b, (short)0, acc, false, false);
    }
    // D: VGPR j -> M=j (lanes 0-15) / M=j+8 (lanes 16-31); N=lane&15.
    // M = local node column, N = batch; +1 skips the phantom node row.
    float* __restrict__ dst = aux + (long)(col0 + (hi ? 8 : 0) + 1) * 16 + m;
#pragma unroll
    for (int j = 0; j < 8; ++j) dst[j * 16] = acc[j];
  }
}

// One edge per 16-lane half-wave; lane = batch index.
__global__ void edge_kernel(const float* __restrict__ aux,
                            const float* __restrict__ param,
                            const int* __restrict__ src,
                            const int* __restrict__ des,
                            float* __restrict__ accum,
                            int E) {
  const int l = threadIdx.x & 15;
  long h       = ((long)blockIdx.x * blockDim.x + threadIdx.x) >> 4;
  long hstride = ((long)gridDim.x * blockDim.x) >> 4;
  for (long e = h; e < E; e += hstride) {
    // Speculative gfx1250 global_prefetch_b8 for the next iteration's
    // metadata (safe out-of-range: speculative prefetches are dropped).
    __builtin_prefetch(src + e + hstride, 0, 1);
    __builtin_prefetch(des + e + hstride, 0, 1);
    __builtin_prefetch(param + e + hstride, 0, 1);

    const int s   = src[e];
    const int d   = des[e];
    const float a = param[e];
    const float w = param[(long)E + e];
    const float b = param[2L * E + e];

    const long soff = (long)s * 16 + l;
    const long doff = (long)d * 16 + l;
    const float vs = aux[soff];
    const float vd = aux[doff];
    const float st = a * tanhf(w * (vs - vd) + b);
    atomicAdd(accum + soff, -st);
    atomicAdd(accum + doff,  st);
  }
}

// accum ((N+1) x 16) -> out (16 x N): out[b][n] = accum[n+1][b].
// Full tiles only; branch-free WMMA path.
__global__ void xpose_bwd_kernel(const float* __restrict__ accum,
                                 float* __restrict__ out,
                                 int N, int ntiles_full) {
  const int lane = threadIdx.x & 31;
  const int wid  = threadIdx.x >> 5;
  const int wpb  = blockDim.x >> 5;
  const int m    = lane & 15;
  const bool hi  = lane >= 16;
  const int k0   = hi ? 2 : 0;

  for (int t = blockIdx.x * wpb + wid; t < ntiles_full; t += gridDim.x * wpb) {
    const int n0 = t << 4;
    const float* __restrict__ srcp = accum + (long)(n0 + 1) * 16 + m;
    v8f acc = {};
#pragma unroll
    for (int p = 0; p < 4; ++p) {
      const int r0 = 4 * p + k0;       // tile rows (nodes) for this A slice
      v2f a;
      a.x = srcp[(long)r0 * 16];
      a.y = srcp[(long)(r0 + 1) * 16];
      v2f b = id_slice(lane, p);
      acc = __builtin_amdgcn_wmma_f32_16x16x4_f32(
          false, a, false, b, (short)0, acc, false, false);
    }
    // D: M = batch (j or j+8), N = local node (lane&15).
    float* __restrict__ dst = out + (long)(hi ? 8 : 0) * N + n0 + m;
#pragma unroll
    for (int j = 0; j < 8; ++j) dst[(long)j * N] = acc[j];
  }
}

// Scalar tail (only launched when N % 16 != 0; never for N=50000).
__global__ void xpose_fwd_tail(const float* __restrict__ x,
                               float* __restrict__ aux, int N, int n0) {
  int i = blockIdx.x * blockDim.x + threadIdx.x;
  int total = (N - n0) * 16;
  if (i < total) {
    int n = n0 + (i >> 4), b = i & 15;
    aux[(long)(n + 1) * 16 + b] = x[(long)b * N + n];
  }
}
__global__ void xpose_bwd_tail(const float* __restrict__ accum,
                               float* __restrict__ out, int N, int n0) {
  int i = blockIdx.x * blockDim.x + threadIdx.x;
  int total = (N - n0) * 16;
  if (i < total) {
    int n = n0 + (i >> 4), b = i & 15;
    out[(long)b * N + n] = accum[(long)(n + 1) * 16 + b];
  }
}

// Fallback if workspace is too small: direct strided gather/scatter.
__global__ void edge_direct_kernel(const float* __restrict__ x,
                                   const float* __restrict__ param,
                                   const int* __restrict__ src,
                                   const int* __restrict__ des,
                                   float* __restrict__ out,
                                   int N, int E) {
  const int l = threadIdx.x & 15;
  long h       = ((long)blockIdx.x * blockDim.x + threadIdx.x) >> 4;
  long hstride = ((long)gridDim.x * blockDim.x) >> 4;
  for (long e = h; e < E; e += hstride) {
    const int s   = src[e];
    const int d   = des[e];
    const float a = param[e];
    const float w = param[(long)E + e];
    const float b = param[2L * E + e];
    const float vs = s ? x[(long)l * N + (s - 1)] : 0.0f;
    const float vd = d ? x[(long)l * N + (d - 1)] : 0.0f;
    const float st = a * tanhf(w * (vs - vd) + b);
    if (s) atomicAdd(out + (long)l * N + (s - 1), -st);
    if (d) atomicAdd(out + (long)l * N + (d - 1),  st);
  }
}

extern "C" void kernel_launch(void* const* d_in, const int* in_sizes, int n_in,
                              void* d_out, int out_size, void* d_ws, size_t ws_size,
                              hipStream_t stream) {
  // setup_inputs order: t(1), x(B*N) f32, param(3E) f32, src(E) i32, des(E) i32
  const float* x     = (const float*)d_in[1];
  const float* param = (const float*)d_in[2];
  const int*   src   = (const int*)d_in[3];
  const int*   des   = (const int*)d_in[4];
  float*       out   = (float*)d_out;

  const int B = 16;                 // batch == WMMA tile dim (layout-hardcoded)
  const int N = in_sizes[1] / B;    // 50000
  const int E = in_sizes[3];        // 1.6M
  const int ntiles_full = N / 16;   // 3125
  const int tail0 = ntiles_full * 16;

  const size_t table_elems = (size_t)(N + 1) * 16;        // (N+1) x 16 floats
  const size_t need = table_elems * 2 * sizeof(float);    // aux + accum ~6.4MB

  if (ws_size >= need) {
    float* aux   = (float*)d_ws;
    float* accum = aux + table_elems;
    // Zero accumulator + aux (covers phantom row 0).
    zero_kernel<<<512, 256, 0, stream>>>((float*)d_ws, (long)table_elems * 2);
    // x -> aux (WMMA transpose), one wave per full tile, 8 waves/block.
    if (ntiles_full > 0)
      xpose_fwd_kernel<<<(ntiles_full + 7) / 8, 256, 0, stream>>>(
          x, aux, N, ntiles_full);
    if (tail0 < N)
      xpose_fwd_tail<<<((N - tail0) * 16 + 255) / 256, 256, 0, stream>>>(
          x, aux, N, tail0);
    // Edge gather / tanh / scatter.
    edge_kernel<<<2048, 256, 0, stream>>>(aux, param, src, des, accum, E);
    // accum -> out (WMMA transpose back, drops phantom row).
    if (ntiles_full > 0)
      xpose_bwd_kernel<<<(ntiles_full + 7) / 8, 256, 0, stream>>>(
          accum, out, N, ntiles_full);
    if (tail0 < N)
      xpose_bwd_tail<<<((N - tail0) * 16 + 255) / 256, 256, 0, stream>>>(
          accum, out, N, tail0);
  } else {
    zero_kernel<<<512, 256, 0, stream>>>(out, (long)B * N);
    edge_direct_kernel<<<2048, 256, 0, stream>>>(x, param, src, des, out, N, E);
  }
}